// TomeMerger_73272142069916
// MI455X (gfx1250) — compile-verified
//
#include <hip/hip_runtime.h>

typedef __attribute__((ext_vector_type(2))) float v2f;
typedef __attribute__((ext_vector_type(8))) float v8f;

#define B_     64
#define T_     64
#define L_     577
#define D_     1024
#define H_     16
#define HGRP   4       // head-groups per batch for the imp reduction
#define NPATCH 576
#define KEEPLEN 288
#define NEWL   290     // 1 cls + 288 kept + 1 merged

// ---------------------------------------------------------------------------
// Kernel 1: partial importance sums, split over 4 head-groups per batch for
// 256 workgroups (5120 waves) of streaming parallelism.
//   partial[b,g,l] = sum_{h in group g} sum_t attn[b,h,t,l]*mask[b,t]
// Lane == l, so every k-step reads 577 contiguous floats (coalesced stream).
// ---------------------------------------------------------------------------
__global__ void imp_partial_kernel(const float* __restrict__ attn,
                                   const float* __restrict__ text_mask,
                                   float* __restrict__ partial) {
    const int b = blockIdx.x >> 2;     // /HGRP
    const int g = blockIdx.x & 3;      // head group: heads 4g..4g+3
    __shared__ float smask[T_];
    if (threadIdx.x < T_) smask[threadIdx.x] = text_mask[b * T_ + threadIdx.x];
    __syncthreads();
    const int l = threadIdx.x;
    if (l >= L_) return;
    const float* base = attn + ((size_t)(b * H_ + g * (H_ / HGRP)) * T_) * L_ + l;
    float acc = 0.f;
    #pragma unroll 8
    for (int k = 0; k < (H_ / HGRP) * T_; ++k) {
        acc = fmaf(base[(size_t)k * L_], smask[k & (T_ - 1)], acc);
    }
    partial[((size_t)b * HGRP + g) * L_ + l] = acc;
}

// ---------------------------------------------------------------------------
// Kernel 2: fold head-group partials + 1/(16*t_len) normalization, then exact
// top-k via stable rank (reproduces jax.lax.top_k ordering: value desc, ties
// by lower index -> rank is a permutation and equals the sorted position).
// Also emits prune weights, deterministic normalizer, new_mask, new_keep_mask.
// One block (576 threads) per batch.
// ---------------------------------------------------------------------------
__global__ void rank_kernel(const float* __restrict__ partial,
                            const float* __restrict__ text_mask,
                            const float* __restrict__ image_mask,
                            const float* __restrict__ prev_keep,
                            float* __restrict__ prune_w,
                            float* __restrict__ inv_norm,
                            int*   __restrict__ sorted_idx,
                            float* __restrict__ out_mask,
                            float* __restrict__ out_keep) {
    const int b = blockIdx.x;
    const int i = threadIdx.x;            // 0..575 over non-cls patches
    __shared__ float s_inv_sh;
    __shared__ float simp[NPATCH];
    if (i == 0) {
        float s = 0.f;
        for (int t = 0; t < T_; ++t) s += text_mask[b * T_ + t];
        s_inv_sh = 1.0f / (16.0f * s);    // t_len >= 1 (mask[:,0]==1)
    }
    float v4 = 0.f;
    #pragma unroll
    for (int g = 0; g < HGRP; ++g) {
        v4 += partial[((size_t)b * HGRP + g) * L_ + 1 + i];
    }
    __syncthreads();
    const float v = v4 * s_inv_sh;        // imp_nc[b, i]
    simp[i] = v;
    __syncthreads();

    int rank = 0;
    #pragma unroll 8
    for (int j = 0; j < NPATCH; ++j) {
        float u = simp[j];
        rank += (int)((u > v) | ((u == v) & (j < i)));
    }
    const bool keep = (rank < KEEPLEN);
    if (keep) sorted_idx[b * KEEPLEN + rank] = i;
    const float pw = keep ? 0.0f : v;
    prune_w[b * NPATCH + i] = pw;

    // new_keep_mask (B,577)
    out_keep[b * L_ + 1 + i] = prev_keep[b * L_ + 1 + i] * (keep ? 1.0f : 0.0f);
    if (i == 0) out_keep[b * L_] = prev_keep[b * L_];

    // new_mask (B,290)
    if (i < NEWL - 1) out_mask[b * NEWL + 1 + i] = 1.0f;
    if (i == 0)       out_mask[b * NEWL]         = image_mask[b * L_];

    // deterministic reduction of prune_w sum (no FP atomics)
    __shared__ float red[NPATCH];
    red[i] = pw;
    __syncthreads();
    if (i < 64) {
        float s = red[i];
        for (int j = i + 64; j < NPATCH; j += 64) s += red[j];
        red[i] = s;
    }
    __syncthreads();
    if (i == 0) {
        float s = 0.f;
        for (int j = 0; j < 64; ++j) s += red[j];
        inv_norm[b] = 1.0f / (s + 1e-6f);
    }
}

// ---------------------------------------------------------------------------
// Kernel 3: new_states rows 0..288 = [cls | gathered kept rows], float4 copies.
// One 256-thread block per output row.
// ---------------------------------------------------------------------------
__global__ void gather_kernel(const float* __restrict__ image_states,
                              const int*   __restrict__ sorted_idx,
                              float*       __restrict__ out_states) {
    const int r = blockIdx.x % (KEEPLEN + 1);   // 0..288
    const int b = blockIdx.x / (KEEPLEN + 1);
    const int src_row = (r == 0) ? 0 : (1 + sorted_idx[b * KEEPLEN + (r - 1)]);
    const float4* src = (const float4*)(image_states + ((size_t)b * L_ + src_row) * D_);
    float4*       dst = (float4*)(out_states + ((size_t)b * NEWL + r) * D_);
    dst[threadIdx.x] = src[threadIdx.x];        // 256 * 16B = 1024 floats
}

// ---------------------------------------------------------------------------
// Kernel 4: merged row via V_WMMA_F32_16X16X4_F32.
// merged[d] = (sum_l w[l] * x[l,d]) * inv_norm, as D(16x16)+=A(16x4)xB(4x16):
//   A: all 16 rows = (w[l0..l0+3])  -> VGPR0 = lane<16 ? w0 : w2,
//                                      VGPR1 = lane<16 ? w1 : w3
//   B: B[k][n] = x[l0+k][d0+n]      -> VGPR0 = rows {0,2}, VGPR1 = rows {1,3}
// Every row of D is identical; lanes 0..15 of c[0] hold the 16 outputs.
// 8 waves/block * 8 blocks cover D=1024 per batch. EXEC all-ones at each WMMA.
// ---------------------------------------------------------------------------
__global__ void merged_wmma_kernel(const float* __restrict__ image_states,
                                   const float* __restrict__ prune_w,
                                   const float* __restrict__ inv_norm,
                                   float* __restrict__ out_states) {
    const int b    = blockIdx.x >> 3;
    const int dblk = blockIdx.x & 7;
    __shared__ float sw[NPATCH];
    for (int i = threadIdx.x; i < NPATCH; i += 256) sw[i] = prune_w[b * NPATCH + i];
    __syncthreads();

    const int wave = threadIdx.x >> 5;
    const int lane = threadIdx.x & 31;
    const int d0   = dblk * 128 + wave * 16;
    const bool lo  = (lane < 16);
    const int rowoff = lo ? 0 : 2;
    // per-lane base: x[b, 1 + rowoff, d0 + lane%16]
    const float* p = image_states + ((size_t)b * L_ + 1 + rowoff) * D_ + d0 + (lane & 15);

    v8f c = {};
    for (int l = 0; l < NPATCH; l += 4) {
        const float w0 = sw[l], w1 = sw[l + 1], w2 = sw[l + 2], w3 = sw[l + 3];
        v2f a, bm;
        a.x  = lo ? w0 : w2;
        a.y  = lo ? w1 : w3;
        bm.x = p[(size_t)l * D_];          // rows l (lanes 0-15) / l+2 (lanes 16-31)
        bm.y = p[(size_t)l * D_ + D_];     // rows l+1 / l+3
        const int lp = (l + 8 < NPATCH) ? (l + 8) : (NPATCH - 4);
        __builtin_prefetch(p + (size_t)lp * D_, 0, 3);   // global_prefetch_b8 (near)
        c = __builtin_amdgcn_wmma_f32_16x16x4_f32(
                /*neg_a=*/false, a, /*neg_b=*/false, bm,
                /*c_mod=*/(short)0, c, /*reuse_a=*/false, /*reuse_b=*/false);
    }
    if (lane < 16) {
        out_states[((size_t)b * NEWL + (NEWL - 1)) * D_ + d0 + lane] = c[0] * inv_norm[b];
    }
}

// ---------------------------------------------------------------------------
extern "C" void kernel_launch(void* const* d_in, const int* in_sizes, int n_in,
                              void* d_out, int out_size, void* d_ws, size_t ws_size,
                              hipStream_t stream) {
    (void)in_sizes; (void)n_in; (void)out_size; (void)ws_size;

    // setup_inputs() order:
    const float* text_states  = (const float*)d_in[0]; (void)text_states; // unused
    const float* text_mask    = (const float*)d_in[1];
    const float* image_states = (const float*)d_in[2];
    const float* image_mask   = (const float*)d_in[3];
    const float* cross_attn   = (const float*)d_in[4];
    const float* prev_keep    = (const float*)d_in[5];
    // d_in[6] = layer_idx (unused)

    // outputs concatenated flat: new_states (64*290*1024), new_mask (64*290),
    // new_keep_mask (64*577), all f32
    float* out_states = (float*)d_out;
    float* out_mask   = out_states + (size_t)B_ * NEWL * D_;
    float* out_keep   = out_mask + (size_t)B_ * NEWL;

    // workspace layout (~750 KB)
    float* partial   = (float*)d_ws;                 // 64*4*577
    float* prune_w   = partial + (size_t)B_ * HGRP * L_;  // 64*576
    float* inv_norm  = prune_w + B_ * NPATCH;        // 64
    int*   sorted_i  = (int*)(inv_norm + B_);        // 64*288

    imp_partial_kernel<<<B_ * HGRP, 640, 0, stream>>>(cross_attn, text_mask, partial);
    rank_kernel<<<B_, NPATCH, 0, stream>>>(partial, text_mask, image_mask, prev_keep,
                                           prune_w, inv_norm, sorted_i,
                                           out_mask, out_keep);
    gather_kernel<<<B_ * (KEEPLEN + 1), 256, 0, stream>>>(image_states, sorted_i, out_states);
    merged_wmma_kernel<<<B_ * 8, 256, 0, stream>>>(image_states, prune_w, inv_norm, out_states);
}